// VnMriReconCell_48223892799740
// MI455X (gfx1250) — compile-verified
//
#include <hip/hip_runtime.h>

// Adaptive spline interpolator: y = lerp(yk[idx,c], yk[idx+1,c], frac)
// over 50.3M elements, channel c = flat_index % 48.
//
// Memory-bound: ~402 MB streamed @ 23.3 TB/s -> ~17 us floor. VALU+DS work
// is ~4x below that, so the design goal is purely clean streaming:
//  - knot table staged to LDS via CDNA5 async global->LDS copy (ASYNCcnt)
//  - repacked as (y_k, y_{k+1}) float2 pairs => one ds_load_b64 per element
//  - 128-bit non-temporal loads/stores, 16 float4/thread, unconditional
//    fast path for full tiles so all 16 loads issue back-to-back.

#define N_FLT    48
#define N_KNOTS  91
#define RAW_ELEMS  (N_FLT * N_KNOTS)        // 4368 floats  (17,472 B)
#define PAIR_ELEMS (N_FLT * (N_KNOTS - 1))  // 4320 float2  (34,560 B)
#define BLOCK 256
#define VPT   16                            // float4s per thread

typedef float v4f __attribute__((ext_vector_type(4)));
typedef float v2f __attribute__((ext_vector_type(2)));
// Exact param type of the async builtin (from hipcc diagnostic):
typedef int i4 __attribute__((vector_size(4 * sizeof(int))));
typedef __attribute__((address_space(1))) i4* gbl_i4_p;
typedef __attribute__((address_space(3))) i4* lds_i4_p;

#if defined(__has_builtin)
#if __has_builtin(__builtin_amdgcn_global_load_async_to_lds_b128) && \
    __has_builtin(__builtin_amdgcn_s_wait_asynccnt)
#define USE_ASYNC_LDS 1
#else
#define USE_ASYNC_LDS 0
#endif
#else
#define USE_ASYNC_LDS 0
#endif

__launch_bounds__(BLOCK)
__global__ void spline_interp_kernel(const float* __restrict__ x,
                                     const float* __restrict__ yk,
                                     float* __restrict__ out,
                                     long long n) {
  __shared__ float s_raw[RAW_ELEMS];
  __shared__ v2f   s_pair[PAIR_ELEMS];

  const int tid = threadIdx.x;

  // ---- Stage 1: raw 17KB knot table -> LDS (CDNA5 async copy, ASYNCcnt) ----
#if USE_ASYNC_LDS
  for (int i = tid; i < RAW_ELEMS / 4; i += BLOCK) {
    __builtin_amdgcn_global_load_async_to_lds_b128(
        (gbl_i4_p)(yk + 4 * i),
        (lds_i4_p)(s_raw + 4 * i),
        /*offset=*/0, /*cpol=*/0);
  }
  __builtin_amdgcn_s_wait_asynccnt(0);   // s_wait_asynccnt 0
#else
  for (int i = tid; i < RAW_ELEMS / 4; i += BLOCK) {
    ((v4f*)s_raw)[i] = ((const v4f*)yk)[i];
  }
#endif
  __syncthreads();

  // ---- Stage 2: paired layout s_pair[k*48+c] = (yk[k,c], yk[k+1,c]) ----
  for (int i = tid; i < PAIR_ELEMS; i += BLOCK) {
    v2f p;
    p.x = s_raw[i];
    p.y = s_raw[i + N_FLT];
    s_pair[i] = p;
  }
  __syncthreads();

  // ---- Stage 3: stream x -> out ----
  const long long nvec = n >> 2;
  const v4f* __restrict__ xv = (const v4f*)x;
  v4f* __restrict__       ov = (v4f*)out;

  const long long tile_base = (long long)blockIdx.x * (BLOCK * VPT);
  const long long base = tile_base + tid;

  // channel of first element of this thread's float4; advances by
  // (BLOCK*4) mod 48 == 16 per iteration (48%4==0: no wrap inside a float4).
  int c0 = (int)((base << 2) % N_FLT);

  auto lerp4 = [&](long long j, int cb) {
    v4f v = __builtin_nontemporal_load(xv + j);     // global_load_b128 th:NT
    v4f o;
#pragma unroll
    for (int e = 0; e < 4; ++e) {
      // xS = (x + 3) / (6/90) = x*15 + 45, clipped to [1e-5, 89.99999]
      float xs = fmaf(v[e], 15.0f, 45.0f);
      xs = fminf(fmaxf(xs, 1e-5f), 89.99999f);
      float xf = floorf(xs);
      float kk = xs - xf;                    // fractional part
      int off = (int)xf * N_FLT + cb + e;    // (int)xf in [0,89] post-clip
      v2f p = s_pair[off];                   // ds_load_b64
      o[e] = fmaf(kk, p.y - p.x, p.x);       // y_f + k*(y_c - y_f)
    }
    __builtin_nontemporal_store(o, ov + j);  // global_store_b128 th:NT
  };

  if (tile_base + (BLOCK * VPT) <= nvec) {
    // Full tile (the only case for the reference shape): no bounds checks,
    // all 16 b128 loads can issue as one clause for max LOADcnt overlap.
#pragma unroll
    for (int it = 0; it < VPT; ++it) {
      lerp4(base + (long long)it * BLOCK, c0);
      c0 += 16;
      if (c0 >= N_FLT) c0 -= N_FLT;
    }
  } else {
#pragma unroll
    for (int it = 0; it < VPT; ++it) {
      const long long j = base + (long long)it * BLOCK;
      if (j < nvec) lerp4(j, c0);
      c0 += 16;
      if (c0 >= N_FLT) c0 -= N_FLT;
    }
    // Scalar tail (n % 4), handled once by the last block (its table is live)
    if ((long long)blockIdx.x == (long long)gridDim.x - 1) {
      for (long long e = (nvec << 2) + tid; e < n; e += BLOCK) {
        float xe = x[e];
        float xs = fminf(fmaxf(fmaf(xe, 15.0f, 45.0f), 1e-5f), 89.99999f);
        float xf = floorf(xs);
        float kk = xs - xf;
        int off = (int)xf * N_FLT + (int)(e % N_FLT);
        v2f p = s_pair[off];
        out[e] = fmaf(kk, p.y - p.x, p.x);
      }
    }
  }
}

extern "C" void kernel_launch(void* const* d_in, const int* in_sizes, int n_in,
                              void* d_out, int out_size, void* d_ws, size_t ws_size,
                              hipStream_t stream) {
  const float* x  = (const float*)d_in[0];   // [16,256,256,48] fp32
  const float* yk = (const float*)d_in[1];   // [91,48] fp32
  float* out = (float*)d_out;

  const long long n = (long long)in_sizes[0];         // 50,331,648
  const long long nvec = n >> 2;
  const long long per_block = (long long)BLOCK * VPT; // 4096 float4s / block
  long long blocks = (nvec + per_block - 1) / per_block;  // 3072, exact
  if (blocks < 1) blocks = 1;                         // tail-only safety

  spline_interp_kernel<<<(int)blocks, BLOCK, 0, stream>>>(x, yk, out, n);
}